// ARIMAX_RNN_11759620456655
// MI455X (gfx1250) — compile-verified
//
#include <hip/hip_runtime.h>

// ============================================================================
// ARIMAX order-4 IIR scan as a blocked WMMA recurrence + TDM tile DMA (gfx1250)
//
//   r[t] = base[t] + c3 r[t-1] + c2 r[t-2] + c1 r[t-3] + c0 r[t-4]
//   base[t] = drift + ma1*y[t] + ma0*y[t-1] + xc*exog[t]
//
// Per 16-step block, per 16-batch tile:
//   R(16x16) = Mb(16x16) @ Base(16x16) + Ms(16x4) @ S(4x16)
//   -> 4x V_WMMA_F32_16X16X4_F32 off the critical path + 1 on it.
// Input tiles (y, exog; 16 rows x 64B, stride T) move via TENSOR_LOAD_TO_LDS
// with hardware row padding 16->20 dwords (conflict-free fragment reads).
// Output tiles leave via TENSOR_STORE_FROM_LDS. All DMA tracked by TENSORcnt.
// Descriptor group1 is loop-invariant (tensor_dim0 = T); only the 64B global
// address bump changes per iteration.
// ============================================================================

typedef __attribute__((ext_vector_type(2))) float    v2f;
typedef __attribute__((ext_vector_type(8))) float    v8f;
typedef __attribute__((ext_vector_type(4))) unsigned v4u;
typedef __attribute__((ext_vector_type(4))) int      v4i;
typedef __attribute__((ext_vector_type(8))) int      v8i;

#define BB    16   // batch tile (N dim)
#define TB    16   // time block (M dim)
#define YSTR  20   // padded LDS row stride (dwords) produced by TDM padding
#define OSTR  16   // output staging stride (TDM store reads LDS contiguously)

__device__ __forceinline__ unsigned lds_off(const void* p) {
    // shared->generic: high 32 bits = LDS aperture, low 32 bits = byte offset
    return (unsigned)(unsigned long long)p;
}

// Loop-invariant D# group1 (ISA ch.8): data_size=4B, tile 16x16, tensor T x 16,
// dim-0 stride = T elements; optional LDS pad 16 dwords -> +4 dwords.
__device__ __forceinline__ v8i tdm_g1(unsigned tdim0, unsigned long long stride,
                                      bool pad)
{
    v8i g1;
    unsigned w0 = 2u << 16;                                 // data_size = 4B
    if (pad) w0 |= (1u << 20)    // pad_enable
               |  (3u << 22)    // pad_interval: every 16 dwords
               |  (3u << 25);   // pad_amount: 4 dwords
    g1[0] = (int)w0;                                        // wg_mask=0 (no cluster)
    g1[1] = (int)((tdim0 & 0xffffu) << 16);                 // tensor_dim0 lo
    g1[2] = (int)(((tdim0 >> 16) & 0xffffu) | (16u << 16)); // dim0 hi | tensor_dim1=16
    g1[3] = (int)(16u << 16);                               // tile_dim0 = 16
    g1[4] = 16;                                             // tile_dim1=16, tile_dim2=0
    g1[5] = (int)(stride & 0xffffffffu);                    // tensor_dim0_stride lo
    g1[6] = (int)((stride >> 32) & 0xffffu);                // stride hi | dim1_stride=0
    g1[7] = 0;
    return g1;
}

// Per-call D# group0: count=1, lds addr, 57-bit global addr, type=2 ("image")
__device__ __forceinline__ v4u tdm_g0(unsigned lds, unsigned long long ga)
{
    v4u g0;
    g0.x = 1u;
    g0.y = lds;
    g0.z = (unsigned)(ga & 0xffffffffu);
    g0.w = (unsigned)((ga >> 32) & 0x01ffffffu) | (2u << 30);
    return g0;
}

__device__ __forceinline__ void tdm_load_tile(unsigned lds,
                                              unsigned long long ga, v8i g1)
{
    v4i z4 = {0, 0, 0, 0};
    v8i z8 = {0, 0, 0, 0, 0, 0, 0, 0};
    __builtin_amdgcn_tensor_load_to_lds(tdm_g0(lds, ga), g1, z4, z4, z8, 0);
}

__device__ __forceinline__ void tdm_store_tile(unsigned lds,
                                               unsigned long long ga, v8i g1)
{
    v4i z4 = {0, 0, 0, 0};
    v8i z8 = {0, 0, 0, 0, 0, 0, 0, 0};
    __builtin_amdgcn_tensor_store_from_lds(tdm_g0(lds, ga), g1, z4, z4, z8, 0);
}

__global__ __launch_bounds__(32) void arimax_wmma_scan(
    const float* __restrict__ y,
    const float* __restrict__ exog,
    const float* __restrict__ ar,     // P+1 = 4, ar[3] == 1
    const float* __restrict__ ma,     // Q = 2
    const float* __restrict__ xcoef,  // 1
    const float* __restrict__ drift,  // 1
    float* __restrict__ out,
    int T)
{
    __shared__ __align__(16) float ytile[2][BB * YSTR]; // TDM-padded y tiles
    __shared__ __align__(16) float etile[2][BB * YSTR]; // TDM-padded exog tiles
    __shared__ __align__(16) float otile[2][BB * OSTR]; // output staging
    __shared__ float hbuf[16];                          // impulse response
    __shared__ float msbuf[64];                         // Ms[i][k] row-major

    const int lane = threadIdx.x;     // 0..31 (wave32)
    const int m    = lane & 15;
    const int hi   = lane >> 4;
    const int n0   = blockIdx.x * BB;

    const float* ygt = y    + (size_t)n0 * T;  // batch-tile base rows
    const float* egt = exog + (size_t)n0 * T;
    float*       ogt = out  + (size_t)n0 * T;

    // Loop-invariant descriptor parts
    const v8i g1L = tdm_g1((unsigned)T, (unsigned long long)T, /*pad=*/true);
    const v8i g1S = tdm_g1((unsigned)T, (unsigned long long)T, /*pad=*/false);
    const unsigned ldsY0 = lds_off(&ytile[0][0]), ldsY1 = lds_off(&ytile[1][0]);
    const unsigned ldsE0 = lds_off(&etile[0][0]), ldsE1 = lds_off(&etile[1][0]);
    const unsigned ldsO0 = lds_off(&otile[0][0]), ldsO1 = lds_off(&otile[1][0]);

    // Kick the TDM immediately: tiles for t = 0 (overlap with setup below)
    tdm_load_tile(ldsY0, (unsigned long long)ygt, g1L);
    tdm_load_tile(ldsE0, (unsigned long long)egt, g1L);

    // c = conv(ar, [-1, 1])[:4]  (D=1 differencing)
    const float c0 = -ar[0];
    const float c1 = ar[0] - ar[1];
    const float c2 = ar[1] - ar[2];
    const float c3 = ar[2] - ar[3];
    const float ma0 = ma[0], ma1 = ma[1];
    const float xc = xcoef[0], dr = drift[0];

    // --- Build Mb (via impulse response h) and Ms once --------------------
    if (lane == 0) {
        float a = 1.f, b = 0.f, cc = 0.f, d = 0.f;
        hbuf[0] = 1.f;
        #pragma unroll
        for (int n = 1; n < 16; ++n) {
            float hn = c3 * a + c2 * b + c1 * cc + c0 * d;
            hbuf[n] = hn;
            d = cc; cc = b; b = a; a = hn;
        }
        #pragma unroll
        for (int k = 0; k < 4; ++k) {
            float s[4] = {0.f, 0.f, 0.f, 0.f};
            s[k] = 1.f;
            #pragma unroll
            for (int i = 0; i < 16; ++i) {
                float r = c0 * s[0] + c1 * s[1] + c2 * s[2] + c3 * s[3];
                msbuf[i * 4 + k] = r;
                s[0] = s[1]; s[1] = s[2]; s[2] = s[3]; s[3] = r;
            }
        }
    }
    __syncthreads();

    // Constant A-fragments (ISA 16x4 f32 A layout: V0=K0|K2, V1=K1|K3)
    v2f aMb[4];
    #pragma unroll
    for (int q = 0; q < 4; ++q) {
        int i0 = m - 4 * q - 2 * hi;          // Mb[m][4q+kk] = h[m-4q-kk]
        aMb[q].x = (i0     >= 0) ? hbuf[i0]     : 0.f;
        aMb[q].y = (i0 - 1 >= 0) ? hbuf[i0 - 1] : 0.f;
    }
    v2f aMs;
    aMs.x = msbuf[m * 4 + 2 * hi];
    aMs.y = msbuf[m * 4 + 2 * hi + 1];

    // First tiles must be resident before fragment reads
    asm volatile("s_wait_tensorcnt 0x0" ::: "memory");

    v2f  sfrag = {0.f, 0.f};  // S fragment (B of Ms@S): V0=K0|K2, V1=K1|K3
    float ym1  = 0.f;         // y[m][t-1] carry (valid in lanes 0..15)

    // Running global addresses: next-tile loads (t+16) and this-tile store (t)
    unsigned long long gaYn = (unsigned long long)ygt + TB * 4ull;
    unsigned long long gaEn = (unsigned long long)egt + TB * 4ull;
    unsigned long long gaO  = (unsigned long long)ogt;

    for (int t = 0; t < T; t += TB) {
        const int cur = (t >> 4) & 1;

        // In-order TENSORcnt: <=1 means both loads for this tile landed and
        // the store issued two iterations ago retired (otile[cur] is free).
        asm volatile("s_wait_tensorcnt 0x1" ::: "memory");

        const float* yt = cur ? &ytile[1][0] : &ytile[0][0];
        const float* et = cur ? &etile[1][0] : &etile[0][0];

        // Fuse base into the B-fragments: bq = base[m][t+4q+2hi .. +1]
        v2f bfrag[4];
        float ytail = 0.f;
        #pragma unroll
        for (int q = 0; q < 4; ++q) {
            const int c = 4 * q + 2 * hi;
            float ym = (q == 0 && hi == 0) ? ym1 : yt[m * YSTR + c - 1];
            v2f yc = *(const v2f*)&yt[m * YSTR + c];
            v2f ec = *(const v2f*)&et[m * YSTR + c];
            bfrag[q].x = fmaf(ma1, yc.x, fmaf(ma0, ym,   fmaf(xc, ec.x, dr)));
            bfrag[q].y = fmaf(ma1, yc.y, fmaf(ma0, yc.x, fmaf(xc, ec.y, dr)));
            if (q == 3) ytail = yc.y;   // lanes 16..31 hold y[m][t+15]
        }

        // Our LDS reads (incl. last iter's from the nxt buffers) must retire
        // before the TDM overwrites those buffers.
        asm volatile("s_wait_dscnt 0x0" ::: "memory");
        if (t + TB < T) {
            tdm_load_tile(cur ? ldsY0 : ldsY1, gaYn, g1L);
            tdm_load_tile(cur ? ldsE0 : ldsE1, gaEn, g1L);
        }

        // Keep far-ahead blocks warming L2 under the TDM traffic
        __builtin_prefetch(ygt + t + 4 * TB, 0, 1);
        __builtin_prefetch(egt + t + 4 * TB, 0, 1);

        // U = Mb @ Base (state-independent), then D = Ms @ S + U (the chain)
        v8f u = {};
        u = __builtin_amdgcn_wmma_f32_16x16x4_f32(false, aMb[0], false,
                bfrag[0], (short)0, u, false, false);
        u = __builtin_amdgcn_wmma_f32_16x16x4_f32(false, aMb[1], false,
                bfrag[1], (short)0, u, false, false);
        u = __builtin_amdgcn_wmma_f32_16x16x4_f32(false, aMb[2], false,
                bfrag[2], (short)0, u, false, false);
        u = __builtin_amdgcn_wmma_f32_16x16x4_f32(false, aMb[3], false,
                bfrag[3], (short)0, u, false, false);
        v8f dmat = __builtin_amdgcn_wmma_f32_16x16x4_f32(false, aMs, false,
                sfrag, (short)0, u, false, false);

        // Stage D as [batch][time] (elem(lane,v) = R[v+8hi][m]); per-lane the
        // 8 dwords are contiguous -> two ds_store_b128. TDM store then walks
        // LDS contiguously and scatters rows with global stride T.
        float* ot = cur ? &otile[1][0] : &otile[0][0];
        #pragma unroll
        for (int v = 0; v < 8; ++v)
            ot[m * OSTR + v + 8 * hi] = dmat[v];

        // Staged tile must be visible to the TDM engine before the store
        asm volatile("s_wait_dscnt 0x0" ::: "memory");
        tdm_store_tile(cur ? ldsO1 : ldsO0, gaO, g1S);

        // Next state S = rows 12..15 of D (all live in lanes 16..31)
        float r12 = __shfl(dmat[4], 16 + m, 32);
        float r13 = __shfl(dmat[5], 16 + m, 32);
        float r14 = __shfl(dmat[6], 16 + m, 32);
        float r15 = __shfl(dmat[7], 16 + m, 32);
        sfrag.x = hi ? r14 : r12;
        sfrag.y = hi ? r15 : r13;

        // Carry y[m][t+15] from lanes 16..31 into lanes 0..15 for next base
        ym1 = __shfl(ytail, 16 + m, 32);

        gaYn += TB * 4ull;
        gaEn += TB * 4ull;
        gaO  += TB * 4ull;
    }
    // implicit S_ENDPGM wait drains the final tensor store
}

extern "C" void kernel_launch(void* const* d_in, const int* in_sizes, int n_in,
                              void* d_out, int out_size, void* d_ws,
                              size_t ws_size, hipStream_t stream)
{
    (void)n_in; (void)d_ws; (void)ws_size; (void)out_size;
    const float* y     = (const float*)d_in[0];
    const float* exog  = (const float*)d_in[1];
    const float* ar    = (const float*)d_in[2];
    const float* ma    = (const float*)d_in[3];
    const float* xcoef = (const float*)d_in[4];
    const float* drift = (const float*)d_in[5];
    float* out = (float*)d_out;

    const int T = 8192;
    const int B = in_sizes[0] / T;   // 1024
    dim3 grid(B / BB);               // 64 batch tiles, one wave32 each
    dim3 block(32);
    arimax_wmma_scan<<<grid, block, 0, stream>>>(y, exog, ar, ma, xcoef, drift,
                                                 out, T);
}